// ChannelWiseTensorProduct_85779086836293
// MI455X (gfx1250) — compile-verified
//
#include <hip/hip_runtime.h>

// ---------------------------------------------------------------------------
// ChannelWiseTensorProduct for MI455X (gfx1250), wave32.
//
// Roofline: ~0.3 GFLOP vs ~500 MB HBM traffic -> memory-bound (~21 us at
// 23.3 TB/s). Per batch row: out_pre(128x11) = X1s(128x4) x M(4x11) where
// M depends only on x2[row] (absorbs 1/sqrt3, 1/sqrt2). Eight
// V_WMMA_F32_16X16X4_F32 per row; weights applied as elementwise post-scale.
//
// This revision software-pipelines the B-operand loads (all 16 issued before
// the WMMA chain so loadcnt waits overlap matrix ops) and removes the
// if/else store divergence in favor of one unconditional b96 store with a
// cndmask-selected address plus a single predicated lo-half block.
// ---------------------------------------------------------------------------

typedef __attribute__((ext_vector_type(2))) float v2f;
typedef __attribute__((ext_vector_type(8))) float v8f;

constexpr int NU      = 128;       // channels
constexpr int ROW_IN  = 4 * NU;    // 512 floats per x1 row
constexpr int ROW_OUT = 11 * NU;   // 1408 floats per output row

// M[i][o]: 4x16 (cols 11..15 zero). i=0 -> scalar irrep s0; i=1..3 -> s1.
__device__ __forceinline__ float m_entry(int i, int o,
                                         float y0, float yx, float yy, float yz) {
  constexpr float R3 = 0.57735026918962576f;  // 1/sqrt(3)
  constexpr float R2 = 0.70710678118654752f;  // 1/sqrt(2)
  if (i == 0) {
    switch (o) {
      case 0: return y0;
      case 1: return yx;
      case 2: return yy;
      case 3: return yz;
      default: return 0.0f;
    }
  }
  const int ii = i - 1;
  switch (o) {
    case 4:  return (ii == 0) ? y0 : 0.0f;
    case 5:  return (ii == 1) ? y0 : 0.0f;
    case 6:  return (ii == 2) ? y0 : 0.0f;
    case 7:  return ((ii == 0) ? yx : (ii == 1) ? yy : yz) * R3;
    case 8:  return (ii == 1) ?  yz * R2 : (ii == 2) ? -yy * R2 : 0.0f;
    case 9:  return (ii == 0) ? -yz * R2 : (ii == 2) ?  yx * R2 : 0.0f;
    case 10: return (ii == 0) ?  yy * R2 : (ii == 1) ? -yx * R2 : 0.0f;
    default: return 0.0f;
  }
}

__global__ __launch_bounds__(256)
void cwtp_wmma_kernel(const float* __restrict__ x1,
                      const float* __restrict__ x2,
                      const float* __restrict__ weight,
                      float* __restrict__ out, int Bn) {
  const int  lane = threadIdx.x & 31;
  const bool hi   = lane >= 16;      // K=2,3 half / D rows 8..15
  const int  col  = lane & 15;
  const int  wave = blockIdx.x * (blockDim.x >> 5) + (threadIdx.x >> 5);
  const int  nwv  = gridDim.x * (blockDim.x >> 5);

  // Row-invariant per-lane weights. wr[t][0] already holds the hi/lo select
  // for the unconditional b96 store (w4 for hi lanes, w1 for lo lanes).
  // wr[t][1..3] = w0,w2,w3 are only read by lo lanes.
  float wr[8][4];
#pragma unroll
  for (int t = 0; t < 8; ++t) {
    const int u = 16 * t + col;
    wr[t][0] = weight[(hi ? 4 : 1) * NU + u];
    wr[t][1] = weight[0 * NU + u];
    wr[t][2] = weight[2 * NU + u];
    wr[t][3] = weight[3 * NU + u];
  }

  for (int row = wave; row < Bn; row += nwv) {   // wave-uniform loop
    const float* __restrict__ x1r  = x1 + (size_t)row * ROW_IN;
    float* __restrict__       outr = out + (size_t)row * ROW_OUT;

    // Prefetch next row's x1 (32 lanes x 64B = 2KB) -> global_prefetch_b8
    if (row + nwv < Bn)
      __builtin_prefetch(x1 + (size_t)(row + nwv) * ROW_IN + lane * 16, 0, 3);

    const float4 y = *reinterpret_cast<const float4*>(x2 + (size_t)row * 4);

    // A = M^T (16x4 f32): lane l -> row o=l%16; VGPR0/1 = K(0,1) lo / K(2,3) hi.
    const int i0 = hi ? 2 : 0, i1 = hi ? 3 : 1;
    v2f a;
    a[0] = m_entry(i0, col, y.x, y.y, y.z, y.w);
    a[1] = m_entry(i1, col, y.x, y.y, y.z, y.w);

    // --- Batch all B-operand loads for the row (software pipelining) ------
    // B[k=i, n=col] = x1s[u,i]; lo lanes need (x1[u], x1[NU+3u]),
    // hi lanes need (x1[NU+3u+1], x1[NU+3u+2]).
    v2f bmv[8];
#pragma unroll
    for (int t = 0; t < 8; ++t) {
      const int u  = 16 * t + col;
      const int s1 = NU + 3 * u;
      bmv[t][0] = x1r[hi ? s1 + 1 : u];
      bmv[t][1] = x1r[hi ? s1 + 2 : s1];
    }

    // --- 8 WMMAs + scaled stores -----------------------------------------
#pragma unroll
    for (int t = 0; t < 8; ++t) {
      const int u  = 16 * t + col;
      const int u3 = 3 * u;

      v8f c = {};
      c = __builtin_amdgcn_wmma_f32_16x16x4_f32(
              /*neg_a=*/false, a, /*neg_b=*/false, bmv[t],
              /*c_mod=*/(short)0, c, /*reuse_a=*/false, /*reuse_b=*/false);

      // Unconditional b96: o1 triple (lo lanes) or o4 triple (hi lanes).
      const int   cb = (hi ? 8 * NU : NU) + u3;
      const float ws = wr[t][0];                  // w1 (lo) / w4 (hi)
      outr[cb]     = (hi ? c[0] : c[1]) * ws;
      outr[cb + 1] = (hi ? c[1] : c[2]) * ws;
      outr[cb + 2] = (hi ? c[2] : c[3]) * ws;

      // Lo-half only: o0, o2 triple, o3. (D rows 11..15 are padding.)
      if (!hi) {
        outr[u]               = c[0] * wr[t][1];  // o0
        outr[4 * NU + u3]     = c[4] * wr[t][2];  // o2_0
        outr[4 * NU + u3 + 1] = c[5] * wr[t][2];  // o2_1
        outr[4 * NU + u3 + 2] = c[6] * wr[t][2];  // o2_2
        outr[7 * NU + u]      = c[7] * wr[t][3];  // o3
      }
    }
  }
}

extern "C" void kernel_launch(void* const* d_in, const int* in_sizes, int n_in,
                              void* d_out, int out_size, void* d_ws, size_t ws_size,
                              hipStream_t stream) {
  const float* x1 = (const float*)d_in[0];   // (B, 512) f32
  const float* x2 = (const float*)d_in[1];   // (B, 4)   f32
  const float* w  = (const float*)d_in[2];   // (1, 640) f32
  float* out      = (float*)d_out;           // (B, 1408) f32

  const int Bn = in_sizes[0] / ROW_IN;       // 65536

  const int threads = 256;                   // 8 waves / block
  int blocks = (Bn + 7) / 8;                 // one row per wave max
  if (blocks > 4096) blocks = 4096;          // grid-stride beyond this

  cwtp_wmma_kernel<<<blocks, threads, 0, stream>>>(x1, x2, w, out, Bn);
}